// VarlenAttention_81372450390387
// MI455X (gfx1250) — compile-verified
//
#include <hip/hip_runtime.h>
#include <math.h>

typedef __attribute__((ext_vector_type(2))) float v2f;
typedef __attribute__((ext_vector_type(8))) float v8f;

#define NH 16
#define HD 64
#define ROWSZ (NH * HD)   // 1024 floats per token row
#define LDS_STRIDE 18     // conflict-free & 8B-aligned for b64 reads

__global__ __launch_bounds__(32)
void va_fwd_kernel(const float* __restrict__ q,
                   const float* __restrict__ k,
                   const float* __restrict__ v,
                   const int*   __restrict__ cuq,
                   const int*   __restrict__ cuk,
                   float*       __restrict__ out)
{
    const int lane = threadIdx.x & 31;
    const int il   = lane & 15;   // row/col index within 16
    const int hi   = lane >> 4;   // K-half selector
    const int b    = blockIdx.y;
    const int h    = blockIdx.z;

    const int cq0  = cuq[b];
    const int cq1  = cuq[b + 1];
    const int lenq = cq1 - cq0;
    const int base = blockIdx.x * 16;
    if (base >= lenq) return;

    const int ck0   = cuk[b];
    const int lenk  = cuk[b + 1] - ck0;
    const int delta = lenk - lenq;
    const int nrows = min(16, lenq - base);

    __shared__ float pl[16 * LDS_STRIDE];

    // ---- load Q A-fragments, pre-scaled by 1/sqrt(D) ----
    int qrow = cq0 + base + il;
    if (qrow > cq1 - 1) qrow = cq1 - 1;            // clamp padded rows (not stored)
    const float* qp = q + (size_t)qrow * ROWSZ + h * HD + 2 * hi;
    v2f qa[16];
#pragma unroll
    for (int c = 0; c < 16; ++c) {
        v2f t = *(const v2f*)(qp + 4 * c);
        qa[c] = t * 0.125f;                        // fold softmax scale into Q
    }

    v8f oacc[4];
#pragma unroll
    for (int nc = 0; nc < 4; ++nc) oacc[nc] = (v8f){};
    v8f lacc = (v8f){};                            // running softmax denominator (all cols equal)
    float mrow[8];
#pragma unroll
    for (int r = 0; r < 8; ++r) mrow[r] = -3.0e38f;

    // causal horizon for this q-tile
    int maxj = base + 15 + delta;
    if (maxj > lenk - 1) maxj = lenk - 1;
    if (maxj < 0) {      // no visible keys: emit zeros for valid rows
#pragma unroll
        for (int r = 0; r < 8; ++r) {
            int m = r + 8 * hi;
            if (m < nrows) {
                size_t ro = (size_t)(cq0 + base + m) * ROWSZ + h * HD;
#pragma unroll
                for (int nc = 0; nc < 4; ++nc) out[ro + nc * 16 + il] = 0.0f;
            }
        }
        return;
    }
    const int ktiles = maxj / 16 + 1;
    const float* kbase = k + (size_t)ck0 * ROWSZ + h * HD;
    const float* vbase = v + (size_t)ck0 * ROWSZ + h * HD;
    const v2f ones = {1.0f, 1.0f};                 // B-fragment of the all-ones 4x16 matrix

    for (int kt = 0; kt < ktiles; ++kt) {
        // prefetch next tile's K/V rows
        if (kt + 1 < ktiles) {
            int pr = min((kt + 1) * 16 + il, lenk - 1);
            __builtin_prefetch(kbase + (size_t)pr * ROWSZ, 0, 3);
            __builtin_prefetch(vbase + (size_t)pr * ROWSZ, 0, 3);
        }

        // ---- S = (Q/sqrt(D)) * K^T : 16 fp32 WMMAs over K-dim 64 ----
        int krow = kt * 16 + il;
        if (krow > lenk - 1) krow = lenk - 1;
        const float* kp = kbase + (size_t)krow * ROWSZ + 2 * hi;
        v8f s = (v8f){};
#pragma unroll
        for (int c = 0; c < 16; ++c) {
            v2f kb = *(const v2f*)(kp + 4 * c);
            s = __builtin_amdgcn_wmma_f32_16x16x4_f32(
                    false, qa[c], false, kb, (short)0, s, false, false);
        }

        // ---- causal/validity mask (C layout: m = r + 8*hi, n = il) ----
        const int jg = kt * 16 + il;
#pragma unroll
        for (int r = 0; r < 8; ++r) {
            int ig = base + r + 8 * hi;
            bool ok = (jg <= ig + delta) && (jg < lenk);
            s[r] = ok ? s[r] : -1.0e9f;
        }

        // ---- online softmax: row max across 16-lane halves ----
        float alpha[8];
#pragma unroll
        for (int r = 0; r < 8; ++r) {
            float x = s[r];
            x = fmaxf(x, __shfl_xor(x, 1, 16));
            x = fmaxf(x, __shfl_xor(x, 2, 16));
            x = fmaxf(x, __shfl_xor(x, 4, 16));
            x = fmaxf(x, __shfl_xor(x, 8, 16));
            float mnew = fmaxf(mrow[r], x);
            s[r]     = __expf(s[r] - mnew);
            alpha[r] = __expf(mrow[r] - mnew);
            mrow[r]  = mnew;
        }
#pragma unroll
        for (int r = 0; r < 8; ++r) {
            oacc[0][r] *= alpha[r]; oacc[1][r] *= alpha[r];
            oacc[2][r] *= alpha[r]; oacc[3][r] *= alpha[r];
            lacc[r]    *= alpha[r];
        }

        // ---- relayout P: C-layout -> A-fragments via LDS ----
#pragma unroll
        for (int r = 0; r < 8; ++r)
            pl[(r + 8 * hi) * LDS_STRIDE + il] = s[r];
        __syncthreads();
        v2f pa[4];
#pragma unroll
        for (int c = 0; c < 4; ++c)
            pa[c] = *(const v2f*)(&pl[il * LDS_STRIDE + 4 * c + 2 * hi]);
        __syncthreads();

        // ---- rowsum(P) via WMMA against all-ones B: l += P * 1 ----
#pragma unroll
        for (int c = 0; c < 4; ++c)
            lacc = __builtin_amdgcn_wmma_f32_16x16x4_f32(
                       false, pa[c], false, ones, (short)0, lacc, false, false);

        // ---- O += P * V : 16 fp32 WMMAs (4 col-chunks x K=16) ----
#pragma unroll
        for (int nc = 0; nc < 4; ++nc) {
#pragma unroll
            for (int c = 0; c < 4; ++c) {
                int kr0 = kt * 16 + 4 * c + 2 * hi;
                int kr1 = kr0 + 1;
                kr0 = min(kr0, lenk - 1);
                kr1 = min(kr1, lenk - 1);
                v2f vb = { vbase[(size_t)kr0 * ROWSZ + nc * 16 + il],
                           vbase[(size_t)kr1 * ROWSZ + nc * 16 + il] };
                oacc[nc] = __builtin_amdgcn_wmma_f32_16x16x4_f32(
                               false, pa[c], false, vb, (short)0, oacc[nc],
                               false, false);
            }
        }
    }

    // ---- normalize and store valid rows ----
#pragma unroll
    for (int r = 0; r < 8; ++r) {
        int m = r + 8 * hi;
        if (m < nrows) {
            float inv = 1.0f / lacc[r];
            size_t ro = (size_t)(cq0 + base + m) * ROWSZ + h * HD;
#pragma unroll
            for (int nc = 0; nc < 4; ++nc)
                out[ro + nc * 16 + il] = oacc[nc][r] * inv;
        }
    }
}

extern "C" void kernel_launch(void* const* d_in, const int* in_sizes, int n_in,
                              void* d_out, int out_size, void* d_ws, size_t ws_size,
                              hipStream_t stream) {
    const float* q   = (const float*)d_in[0];
    const float* k   = (const float*)d_in[1];
    const float* v   = (const float*)d_in[2];
    const int*   cuq = (const int*)d_in[3];
    const int*   cuk = (const int*)d_in[4];
    float* out = (float*)d_out;

    const int T = in_sizes[0] / ROWSZ;       // total tokens
    const int B = in_sizes[3] - 1;           // batches
    dim3 grid((T + 15) / 16, B, NH);
    va_fwd_kernel<<<grid, 32, 0, stream>>>(q, k, v, cuq, cuk, out);
}